// RBFInterpolationKernel_74947179315854
// MI455X (gfx1250) — compile-verified
//
#include <hip/hip_runtime.h>

typedef float v2f __attribute__((ext_vector_type(2)));
typedef float v8f __attribute__((ext_vector_type(8)));

#define BATCH 4
#define NPTS  256
#define MPTS  65536
// exp(-d2/(2*sigma^2)) = exp2(d2 * S),  S = -log2(e)/(2*0.1^2)
#define SCALE_S (-72.13475204444817f)

__global__ __launch_bounds__(256) void rbf_wmma_kernel(
    const float* __restrict__ sparse,   // [B, N, 3]
    const float* __restrict__ grid,     // [M, 3]
    float* __restrict__ out)            // [B, M, N]
{
    // Two K-pair planes: spk[0][n] = {sx, sy} (K0,K1), spk[1][n] = {sz, s^2} (K2,K3)
    __shared__ float2 spk[2][NPTS];
    __shared__ __align__(32) float g2buf[8 * 16];   // per-wave S*|g|^2 per row

    const int tid    = threadIdx.x;
    const int wave   = tid >> 5;
    const int lane   = tid & 31;
    const int lanelo = lane & 15;
    const int h      = lane >> 4;       // half-wave: K-pair select / M-offset

    const int b     = blockIdx.y;
    const int mbase = (blockIdx.x * 8 + wave) * 16;

    // ---- stage sparse batch into LDS, split by K-pair ----
    {
        const float* s = sparse + ((size_t)b * NPTS + tid) * 3;
        float sx = s[0], sy = s[1], sz = s[2];
        spk[0][tid] = make_float2(sx, sy);
        spk[1][tid] = make_float2(sz, sx * sx + sy * sy + sz * sz);
    }

    // ---- per-lane grid row coords (both halves read the same 16 rows) ----
    const float* g = grid + (size_t)(mbase + lanelo) * 3;
    const float gx = g[0], gy = g[1], gz = g[2];
    if (lane < 16)
        g2buf[wave * 16 + lane] = SCALE_S * (gx * gx + gy * gy + gz * gz);

    __syncthreads();

    // A row = S * [-2gx, -2gy, -2gz, 1]; with C = S*g^2 tile:
    //   D = S*(-2 g.s + s^2 + g^2) = S * d2  (exp2 argument, no post-scale)
    v2f Amat;
    {
        const float m2s = -2.0f * SCALE_S;
        Amat.x = h ? (m2s * gz) : (m2s * gx);
        Amat.y = h ? SCALE_S    : (m2s * gy);
    }

    // C tile: slot r, half h -> S*g^2 of row (mbase + r + 8h)
    v8f cg2 = *(const v8f*)&g2buf[wave * 16 + h * 8];

    const float2* bp = &spk[h][lanelo];

    float w[16][8];     // all 256 weights for this lane's rows, kept in VGPRs
    float acc[8];
    #pragma unroll
    for (int r = 0; r < 8; ++r) acc[r] = 0.0f;

    #pragma unroll
    for (int t = 0; t < 16; ++t) {
        float2 s = bp[t * 16];                  // ds_load_b64 (pairs into 2addr)
        v2f Bmat; Bmat.x = s.x; Bmat.y = s.y;

        v8f d = __builtin_amdgcn_wmma_f32_16x16x4_f32(
            /*neg_a=*/false, Amat, /*neg_b=*/false, Bmat,
            /*c_mod=*/(short)0, cg2, /*reuse_a=*/false, /*reuse_b=*/false);

        #pragma unroll
        for (int r = 0; r < 8; ++r) {
            // arg = min(S*d2, 0): raw v_min_num_f32, no canonicalize
            // (WMMA output is never sNaN, so IEEE quieting is unnecessary)
            float arg;
            asm("v_min_num_f32 %0, 0, %1" : "=v"(arg) : "v"(d[r]));
            float e = __builtin_amdgcn_exp2f(arg);
            w[t][r] = e;
            acc[r] += e;
        }
    }

    // ---- row sums over N: butterfly within each 16-lane half ----
    #pragma unroll
    for (int r = 0; r < 8; ++r) {
        float a = acc[r];
        a += __shfl_xor(a, 1, 32);
        a += __shfl_xor(a, 2, 32);
        a += __shfl_xor(a, 4, 32);
        a += __shfl_xor(a, 8, 32);
        acc[r] = 1.0f / (a + 1e-8f);
    }

    // ---- normalized store: half-wave writes 64 contiguous bytes per (t,r) ----
    float* o = out + (((size_t)b * MPTS + mbase + 8 * h) * NPTS) + lanelo;
    #pragma unroll
    for (int t = 0; t < 16; ++t) {
        #pragma unroll
        for (int r = 0; r < 8; ++r) {
            o[(size_t)r * NPTS + t * 16] = w[t][r] * acc[r];
        }
    }
}

extern "C" void kernel_launch(void* const* d_in, const int* in_sizes, int n_in,
                              void* d_out, int out_size, void* d_ws, size_t ws_size,
                              hipStream_t stream) {
    const float* sparse = (const float*)d_in[0];   // [4, 256, 3]
    const float* grid   = (const float*)d_in[1];   // [65536, 3]
    float* out          = (float*)d_out;           // [4, 65536, 256]

    dim3 gdim(MPTS / (8 * 16), BATCH, 1);          // (512, 4): 8 waves x 16 rows
    rbf_wmma_kernel<<<gdim, dim3(256, 1, 1), 0, stream>>>(sparse, grid, out);
}